// My_GNN_GNN_NN_80711025427086
// MI455X (gfx1250) — compile-verified
//
#include <hip/hip_runtime.h>
#include <hip/hip_bf16.h>

typedef __attribute__((ext_vector_type(2))) float v2f;
typedef __attribute__((ext_vector_type(8))) float v8f;

#define GCN_N 200000
#define GCN_E 6400000
#define GCN_L 10

// ---------------------------------------------------------------------------
// deg[i] = 1 (self loop)
__global__ void gcn_init_deg(float* __restrict__ deg, int n) {
    int i = blockIdx.x * blockDim.x + threadIdx.x;
    if (i < n) deg[i] = 1.0f;
}

// deg[dst[e]] += 1
__global__ void gcn_count_deg(const long long* __restrict__ dst,
                              float* __restrict__ deg, int e) {
    int i = blockIdx.x * blockDim.x + threadIdx.x;
    if (i < e) {
        int d = (int)dst[i];
        (void)__hip_atomic_fetch_add(&deg[d], 1.0f, __ATOMIC_RELAXED,
                                     __HIP_MEMORY_SCOPE_AGENT);
    }
}

// dis[i] = rsqrt(deg[i])   (in place; deg >= 1 always)
__global__ void gcn_make_dis(float* __restrict__ deg_dis, int n) {
    int i = blockIdx.x * blockDim.x + threadIdx.x;
    if (i < n) deg_dis[i] = rsqrtf(deg_dis[i]);
}

// ---------------------------------------------------------------------------
// Packed edge build (one-time int64 -> int32 decode + norm fold):
//   sp[e] = { src32, bitcast(norm) }  (8 B, one b64 load per edge per layer)
//   dp[e] = dst32                     (4 B)
__global__ void gcn_build_edges(const long long* __restrict__ src,
                                const long long* __restrict__ dst,
                                const float* __restrict__ dis,
                                int2* __restrict__ sp,
                                int* __restrict__ dp, int e) {
    int i = blockIdx.x * blockDim.x + threadIdx.x;
    if (i < e) {
        int s = (int)src[i];
        int d = (int)dst[i];
        float n = dis[s] * dis[d];
        sp[i] = make_int2(s, __float_as_int(n));
        dp[i] = d;
    }
}

// Fallback (small-workspace) path: norm[e] only, indices re-read as int64.
__global__ void gcn_make_norm(const long long* __restrict__ src,
                              const long long* __restrict__ dst,
                              const float* __restrict__ dis,
                              float* __restrict__ norm, int e) {
    int i = blockIdx.x * blockDim.x + threadIdx.x;
    if (i < e) {
        norm[i] = dis[(int)src[i]] * dis[(int)dst[i]];
    }
}

// ---------------------------------------------------------------------------
// Per-layer node transform via V_WMMA_F32_16X16X4_F32.
// One wave handles a tile of 16 nodes:
//   A (16x4 f32): lanes 0-15 hold node features (K=0 -> a.x, K=1 -> a.y),
//                 lanes 16-31 (K=2,3) are zero padding.
//   B (4x16 f32): W in the top-left 2x2; row K=0 in b.x / K=1 in b.y striped
//                 across lanes 0-15; lanes 16-31 (K=2,3 rows) zero.
//   D (16x16 f32, 8 VGPRs): lane L<16 VGPR v = D[v][L];
//                           lane L>=16 VGPR v = D[8+v][L-16].
// Only columns 0,1 of D are meaningful -> lanes 0,1,16,17 write results.
// Also initializes out[i] = b + t[i] * dis[i]^2 (self-loop term + bias).
__global__ void gcn_node_wmma(const float2* __restrict__ h,
                              const float* __restrict__ Wl,  // 4 floats, row-major [2][2]
                              const float* __restrict__ bl,  // 2 floats
                              const float* __restrict__ dis,
                              float2* __restrict__ t,
                              float2* __restrict__ out,
                              int ntiles) {
    int gid   = blockIdx.x * blockDim.x + threadIdx.x;
    int wave  = gid >> 5;
    int lane  = threadIdx.x & 31;
    int nwave = (gridDim.x * blockDim.x) >> 5;

    float w00 = Wl[0], w01 = Wl[1], w10 = Wl[2], w11 = Wl[3];
    float b0 = bl[0], b1 = bl[1];

    v2f bm = {0.0f, 0.0f};
    if (lane == 0)      { bm.x = w00; bm.y = w10; }  // column 0 of W
    else if (lane == 1) { bm.x = w01; bm.y = w11; }  // column 1 of W

    for (int tile = wave; tile < ntiles; tile += nwave) {
        int base = tile * 16;

        v2f a = {0.0f, 0.0f};
        if (lane < 16) {
            float2 hv = h[base + lane];
            a.x = hv.x;   // K = 0
            a.y = hv.y;   // K = 1
        }

        v8f c = {};
        c = __builtin_amdgcn_wmma_f32_16x16x4_f32(
                /*neg_a=*/false, a, /*neg_b=*/false, bm,
                /*c_mod=*/(short)0, c, /*reuse_a=*/false, /*reuse_b=*/false);

        int col  = lane & 15;          // output column (feature index)
        int roff = (lane & 16) >> 1;   // 0 for lanes<16, 8 for lanes>=16
        if (col < 2) {
            #pragma unroll
            for (int v = 0; v < 8; ++v) {
                int node = base + roff + v;
                float tv = c[v];
                float dn = dis[node];
                float sn = dn * dn;                       // self-loop norm = 1/deg
                ((float*)&t[node])[col]   = tv;
                ((float*)&out[node])[col] =
                    ((col == 0) ? b0 : b1) + tv * sn;
            }
        }
    }
}

// ---------------------------------------------------------------------------
// Packed path: out[dp[e]] += t[sp[e].x] * sp[e].norm  (12 B/edge streamed)
__global__ void gcn_edge_scatter_packed(const int2* __restrict__ sp,
                                        const int* __restrict__ dp,
                                        const float2* __restrict__ t,
                                        float2* __restrict__ out, int e) {
    int i = blockIdx.x * blockDim.x + threadIdx.x;
    if (i < e) {
        int2 p = sp[i];
        int s = p.x;
        float n = __int_as_float(p.y);
        int d = dp[i];
        float2 v = t[s];
        float* op = (float*)&out[d];
        (void)__hip_atomic_fetch_add(&op[0], v.x * n, __ATOMIC_RELAXED,
                                     __HIP_MEMORY_SCOPE_AGENT);
        (void)__hip_atomic_fetch_add(&op[1], v.y * n, __ATOMIC_RELAXED,
                                     __HIP_MEMORY_SCOPE_AGENT);
    }
}

// Fallback path: re-read int64 indices each layer (20 B/edge streamed)
__global__ void gcn_edge_scatter(const long long* __restrict__ src,
                                 const long long* __restrict__ dst,
                                 const float* __restrict__ norm,
                                 const float2* __restrict__ t,
                                 float2* __restrict__ out, int e) {
    int i = blockIdx.x * blockDim.x + threadIdx.x;
    if (i < e) {
        int s = (int)src[i];
        int d = (int)dst[i];
        float n = norm[i];
        float2 v = t[s];
        float* op = (float*)&out[d];
        (void)__hip_atomic_fetch_add(&op[0], v.x * n, __ATOMIC_RELAXED,
                                     __HIP_MEMORY_SCOPE_AGENT);
        (void)__hip_atomic_fetch_add(&op[1], v.y * n, __ATOMIC_RELAXED,
                                     __HIP_MEMORY_SCOPE_AGENT);
    }
}

// ---------------------------------------------------------------------------
// d_out[f*N + i] = h[i].f   (reference returns h.T, shape [FEAT, N])
__global__ void gcn_transpose_out(const float2* __restrict__ h,
                                  float* __restrict__ out, int n) {
    int i = blockIdx.x * blockDim.x + threadIdx.x;
    if (i < n) {
        float2 v = h[i];
        out[i]     = v.x;
        out[n + i] = v.y;
    }
}

// ---------------------------------------------------------------------------
extern "C" void kernel_launch(void* const* d_in, const int* in_sizes, int n_in,
                              void* d_out, int out_size, void* d_ws, size_t ws_size,
                              hipStream_t stream) {
    const float*     x   = (const float*)d_in[0];            // [N,2]
    const long long* ei  = (const long long*)d_in[1];        // [2,E] int64
    const float*     Ws  = (const float*)d_in[2];            // [10,2,2]
    const float*     bs  = (const float*)d_in[3];            // [10,2]
    float*           out = (float*)d_out;                    // [2,N]

    const long long* src = ei;
    const long long* dst = ei + GCN_E;

    // Workspace layout (bytes):
    //   dis : N floats           @ 0
    //   t   : N float2           @ 4N
    //   hA  : N float2           @ 12N
    //   hB  : N float2           @ 20N
    //   edge data                @ 28N
    //     packed  : sp (E int2) + dp (E int)  -> 12E bytes
    //     fallback: norm (E floats)           ->  4E bytes
    char* ws = (char*)d_ws;
    float*  dis  = (float*)(ws);
    float2* t    = (float2*)(ws + (size_t)GCN_N * 4);
    float2* hA   = (float2*)(ws + (size_t)GCN_N * 12);
    float2* hB   = (float2*)(ws + (size_t)GCN_N * 20);
    char*   ebase = ws + (size_t)GCN_N * 28;

    const size_t need_packed = (size_t)GCN_N * 28 + (size_t)GCN_E * 12;
    const bool packed = (ws_size >= need_packed);

    int2*  sp   = (int2*)ebase;
    int*   dp   = (int*)(ebase + (size_t)GCN_E * 8);
    float* norm = (float*)ebase;
    (void)in_sizes; (void)n_in; (void)out_size;

    const int TB = 256;
    const int nodeBlocks = (GCN_N + TB - 1) / TB;   // 782
    const int edgeBlocks = (GCN_E + TB - 1) / TB;   // 25000
    const int ntiles     = GCN_N / 16;              // 12500 (exact)

    // Normalization setup (recomputed every call: deterministic, no state).
    gcn_init_deg<<<nodeBlocks, TB, 0, stream>>>(dis, GCN_N);
    gcn_count_deg<<<edgeBlocks, TB, 0, stream>>>(dst, dis, GCN_E);
    gcn_make_dis<<<nodeBlocks, TB, 0, stream>>>(dis, GCN_N);
    if (packed) {
        gcn_build_edges<<<edgeBlocks, TB, 0, stream>>>(src, dst, dis, sp, dp,
                                                       GCN_E);
    } else {
        gcn_make_norm<<<edgeBlocks, TB, 0, stream>>>(src, dst, dis, norm,
                                                     GCN_E);
    }

    // h0 = x
    hipMemcpyAsync(hA, x, (size_t)GCN_N * 2 * sizeof(float),
                   hipMemcpyDeviceToDevice, stream);

    float2* cur = hA;
    float2* nxt = hB;
    for (int l = 0; l < GCN_L; ++l) {
        gcn_node_wmma<<<256, TB, 0, stream>>>(cur, Ws + 4 * l, bs + 2 * l,
                                              dis, t, nxt, ntiles);
        if (packed) {
            gcn_edge_scatter_packed<<<edgeBlocks, TB, 0, stream>>>(sp, dp, t,
                                                                   nxt, GCN_E);
        } else {
            gcn_edge_scatter<<<edgeBlocks, TB, 0, stream>>>(src, dst, norm, t,
                                                            nxt, GCN_E);
        }
        float2* tmp = cur; cur = nxt; nxt = tmp;
    }

    gcn_transpose_out<<<nodeBlocks, TB, 0, stream>>>(cur, out, GCN_N);
}